// CrossGraphAttention_67997922230512
// MI455X (gfx1250) — compile-verified
//
#include <hip/hip_runtime.h>

typedef __attribute__((ext_vector_type(2))) float v2f;
typedef __attribute__((ext_vector_type(8))) float v8f;

constexpr int BATCH = 2, L = 512, DN = 256, DE = 64, H = 8, DK = 32;
constexpr float SLOPE = 0.2f;

__device__ __forceinline__ float leaky(float x) { return x >= 0.f ? x : SLOPE * x; }

__device__ __forceinline__ v8f wmma4(v2f a, v2f b, v8f c) {
  // V_WMMA_F32_16X16X4_F32 : D = A(16x4,f32) * B(4x16,f32) + C(16x16,f32)
  return __builtin_amdgcn_wmma_f32_16x16x4_f32(false, a, false, b, (short)0, c, false, false);
}

// Async DMA global -> LDS, 128-bit per lane, tracked by ASYNCcnt.
__device__ __forceinline__ void async_load_b128(const float* gsrc, float* ldst) {
  auto l3 = (__attribute__((address_space(3))) float*)ldst;
  asm volatile("global_load_async_to_lds_b128 %0, %1, off"
               :
               : "v"(l3), "v"(gsrc)
               : "memory");
}
// Async DMA LDS -> global, 128-bit per lane, tracked by ASYNCcnt.
__device__ __forceinline__ void async_store_b128(float* gdst, const float* lsrc) {
  auto l3 = (__attribute__((address_space(3))) const float*)lsrc;
  asm volatile("global_store_async_from_lds_b128 %0, %1, off"
               :
               : "v"(gdst), "v"(l3)
               : "memory");
}
__device__ __forceinline__ void wait_async0() {
  asm volatile("s_wait_asynccnt 0" ::: "memory");
}
__device__ __forceinline__ void wait_ds0() {
  asm volatile("s_wait_dscnt 0" ::: "memory");
}

// ---- workspace layout (float offsets) ----
constexpr int WS_MFOLD = 0;                        // 64x64 folded edge matrix
constexpr int WS_CW    = WS_MFOLD + 64 * 64;       // 64   folded constant vec
constexpr int WS_AT    = WS_CW + 64;               // 64x8 logit matrix (target)
constexpr int WS_AB    = WS_AT + 512;              // 64x8 logit matrix (binder)
constexpr int WS_CT    = WS_AB + 512;              // 8
constexpr int WS_CB    = WS_CT + 8;                // 8
constexpr int WS_TIW   = WS_CB + 8;                // (B*L,64) per-i folded vec
constexpr int WS_BJW   = WS_TIW + BATCH * L * 64;  // (B*L,64) per-j folded vec
constexpr int WS_TP2   = WS_BJW + BATCH * L * 64;  // (B*L,256)
constexpr int WS_BP2   = WS_TP2 + BATCH * L * DN;
constexpr int WS_WHT   = WS_BP2 + BATCH * L * DN;  // tp2 @ Wh
constexpr int WS_WHB   = WS_WHT + BATCH * L * DN;  // bp2 @ Wh
constexpr int WS_QT    = WS_WHB + BATCH * L * DN;  // (B*L,8)
constexpr int WS_KT    = WS_QT + BATCH * L * H;
constexpr int WS_QB    = WS_KT + BATCH * L * H;
constexpr int WS_KB    = WS_QB + BATCH * L * H;
constexpr int WS_AGGT  = WS_KB + BATCH * L * H;    // (B*L,256)
constexpr int WS_AGGB  = WS_AGGT + BATCH * L * DN;
constexpr int WS_SET   = WS_AGGB + BATCH * L * DN; // (B,L1,L2,H) edge logits t
constexpr int WS_SEB   = WS_SET + BATCH * L * L * H; // (B,L2,L1,H) edge logits b

// ============ K0: fold small matrices ============
__global__ __launch_bounds__(64) void setup_kernel(
    const float* __restrict__ We_w, const float* __restrict__ We_b,
    const float* __restrict__ We2_w, const float* __restrict__ We2_b,
    const float* __restrict__ edge_w, const float* __restrict__ edge_b,
    const float* __restrict__ out_e_w, const float* __restrict__ out_e_b,
    const float* __restrict__ attn_t_w, const float* __restrict__ attn_b_w,
    float* __restrict__ ws) {
  __shared__ float M1[64 * 64];
  __shared__ float c0[64];
  const int r = threadIdx.x;
  // M1 = We_w @ blockdiag(Ee);  Ee = edge_w rows 0..7
  for (int c = 0; c < 64; ++c) {
    int h = c >> 3, f = c & 7;
    float s = 0.f;
    for (int e = 0; e < 8; ++e) s += We_w[r * 64 + h * 8 + e] * edge_w[e * 8 + f];
    M1[r * 64 + c] = s;
  }
  {
    int h = r >> 3, f = r & 7;
    float s = edge_b[f];
    for (int e = 0; e < 8; ++e) s += We_b[h * 8 + e] * edge_w[e * 8 + f];
    c0[r] = s;
  }
  __syncthreads();
  for (int c = 0; c < 64; ++c) {
    float s = 0.f;
    for (int k = 0; k < 64; ++k) s += M1[r * 64 + k] * out_e_w[k * 64 + c];
    ws[WS_MFOLD + r * 64 + c] = s;
  }
  {
    float s = out_e_b[r];
    for (int k = 0; k < 64; ++k) s += c0[k] * out_e_w[k * 64 + r];
    ws[WS_CW + r] = s;
  }
  // A_t/A_b: fold We2 with the 8-dim edge slice of the attention vectors
  for (int h = 0; h < 8; ++h) {
    float st = 0.f, sb = 0.f;
    for (int e = 0; e < 8; ++e) {
      float w = We2_w[r * 64 + h * 8 + e];
      st += w * attn_t_w[64 + e];
      sb += w * attn_b_w[64 + e];
    }
    ws[WS_AT + r * 8 + h] = st;
    ws[WS_AB + r * 8 + h] = sb;
  }
  if (r < 8) {
    float st = 0.f, sb = 0.f;
    for (int e = 0; e < 8; ++e) {
      st += We2_b[r * 8 + e] * attn_t_w[64 + e];
      sb += We2_b[r * 8 + e] * attn_b_w[64 + e];
    }
    ws[WS_CT + r] = st;
    ws[WS_CB + r] = sb;
  }
}

// ============ K1: per-row node projections ============
__global__ __launch_bounds__(256) void node_kernel(
    const float* __restrict__ target, const float* __restrict__ binder,
    const float* __restrict__ diff,
    const float* __restrict__ Wn_w, const float* __restrict__ Wn_b,
    const float* __restrict__ Wn2_w, const float* __restrict__ Wn2_b,
    const float* __restrict__ Wd_w, const float* __restrict__ Wd_b,
    const float* __restrict__ Wd2_w, const float* __restrict__ Wd2_b,
    const float* __restrict__ Wh_w, const float* __restrict__ Wh_b,
    const float* __restrict__ edge_w, const float* __restrict__ out_e_w,
    const float* __restrict__ attn_t_w, const float* __restrict__ attn_b_w,
    float* __restrict__ ws) {
  const bool isT = blockIdx.x < (unsigned)(BATCH * L);
  const int r = isT ? blockIdx.x : blockIdx.x - BATCH * L;
  const int tid = threadIdx.x;
  __shared__ float x[256], dvec[64], proj[256], proj2[256], dps[64], dp2s[64], f64[64];
  x[tid] = (isT ? target : binder)[r * DN + tid];
  if (isT && tid < 64) dvec[tid] = diff[r * 64 + tid];
  __syncthreads();
  float a1 = Wn_b[tid], a2 = Wn2_b[tid];
  for (int k = 0; k < DN; ++k) {
    float xv = x[k];
    a1 += xv * Wn_w[k * DN + tid];
    a2 += xv * Wn2_w[k * DN + tid];
  }
  proj[tid] = a1;
  proj2[tid] = a2;
  ws[(isT ? WS_TP2 : WS_BP2) + r * DN + tid] = a2;
  if (isT && tid < 64) {
    float d1 = Wd_b[tid], d2 = Wd2_b[tid];
    for (int k = 0; k < 64; ++k) {
      float dv = dvec[k];
      d1 += dv * Wd_w[k * 64 + tid];
      d2 += dv * Wd2_w[k * 64 + tid];
    }
    dps[tid] = d1;
    dp2s[tid] = d2;
  }
  __syncthreads();
  if (tid < 64) {  // Ti64 = tp@Ei + dp@Ed   (or Bj64 = bp@Ej)
    int h = tid >> 3, e = tid & 7;
    float s = 0.f;
    const float* E = edge_w + (isT ? 8 : 40) * 8;
    for (int k = 0; k < 32; ++k) s += proj[h * 32 + k] * E[k * 8 + e];
    if (isT) {
      const float* Ed = edge_w + 72 * 8;
      for (int k = 0; k < 8; ++k) s += dps[h * 8 + k] * Ed[k * 8 + e];
    }
    f64[tid] = s;
  }
  __syncthreads();
  if (tid < 64) {  // fold through out_e_w
    float s = 0.f;
    for (int k = 0; k < 64; ++k) s += f64[k] * out_e_w[k * 64 + tid];
    ws[(isT ? WS_TIW : WS_BJW) + r * 64 + tid] = s;
  }
  {  // per-head Wh projection of proj2
    int h = tid >> 5, k = tid & 31;
    float s = Wh_b[k];
    for (int m = 0; m < 32; ++m) s += proj2[h * 32 + m] * Wh_w[m * 32 + k];
    ws[(isT ? WS_WHT : WS_WHB) + r * DN + tid] = s;
  }
  if (tid < H) {  // attention row/col scalars
    if (isT) {
      float s = 0.f;
      for (int k = 0; k < 32; ++k) s += proj2[tid * 32 + k] * attn_t_w[k];
      for (int k = 0; k < 8; ++k) s += dp2s[tid * 8 + k] * attn_t_w[72 + k];
      ws[WS_QT + r * H + tid] = s;
      float s2 = 0.f;
      for (int k = 0; k < 32; ++k) s2 += proj2[tid * 32 + k] * attn_b_w[32 + k];
      ws[WS_KB + r * H + tid] = s2;
    } else {
      float s = 0.f;
      for (int k = 0; k < 32; ++k) s += proj2[tid * 32 + k] * attn_b_w[k];
      ws[WS_QB + r * H + tid] = s;
      float s2 = 0.f;
      for (int k = 0; k < 32; ++k) s2 += proj2[tid * 32 + k] * attn_t_w[32 + k];
      ws[WS_KT + r * H + tid] = s2;
    }
  }
}

// ============ K2: folded edge GEMM (WMMA) + fused logit matvecs ============
// Each wave owns one 16x64 edge tile (a contiguous 4 KB block):
//   in : global_load_async_to_lds_b128 (overlapped with Mfold staging+barrier)
//   mma: A-frags from LDS (pitch 68 -> conflict-free b64), B-frags from a
//        pair-interleaved Mfold layout (one aligned ds_load_b64 per frag)
//   out: tile staged to LDS, then global_store_async_from_lds_b128 (full
//        512B-contiguous DMA stores), while the wave runs the logit matvecs.
__global__ __launch_bounds__(256) void edge_kernel(
    const float* __restrict__ edge, float* __restrict__ ws,
    float* __restrict__ new_edge) {
  __shared__ float ldsM2[32 * 130]; // Mfold rows (2r,2r+1) interleaved per col
  __shared__ float ldsAt[512], ldsAb[512], ldsCw[64], ldsCt[8], ldsCb[8];
  __shared__ float outst[8][16 * 68];   // per-wave output tile, pitch 68
  __shared__ float tilebuf[8][16 * 68]; // per-wave edge tile, pitch 68
  const int tid = threadIdx.x;
  const int wave = tid >> 5, lane = tid & 31;
  const int half = lane >> 4, cl = lane & 15;
  const int tile = blockIdx.x * 8 + wave;  // 32768 tiles total
  const int b = tile >> 14;                // 16384 tiles per batch
  const int rem = tile & 16383;
  const int i = rem >> 5;
  const int j0 = (rem & 31) << 4;

  // kick off the tile DMA first: 8 x b128 per lane-group, fully coalesced
  const float* tbase = edge + (((size_t)(b * L + i)) * L + j0) * DE;
  float* mybuf = &tilebuf[wave][0];
#pragma unroll
  for (int t = 0; t < 8; ++t) {
    const int idx = t * 32 + lane;          // 16-byte chunk id within tile
    const int row = idx >> 4, c4 = idx & 15;
    async_load_b128(tbase + idx * 4, mybuf + row * 68 + c4 * 4);
  }

  // overlap: cooperative staging of folded matrices while the DMA flies
  for (int idx = tid; idx < 4096; idx += 256) {
    int rr = idx >> 6, cc = idx & 63;
    ldsM2[(rr >> 1) * 130 + 2 * cc + (rr & 1)] = ws[WS_MFOLD + idx];
  }
  for (int idx = tid; idx < 512; idx += 256) {
    ldsAt[idx] = ws[WS_AT + idx];
    ldsAb[idx] = ws[WS_AB + idx];
  }
  if (tid < 64) ldsCw[tid] = ws[WS_CW + tid];
  if (tid < 8) {
    ldsCt[tid] = ws[WS_CT + tid];
    ldsCb[tid] = ws[WS_CB + tid];
  }
  __syncthreads();
  wait_async0();  // our wave's tile is resident in LDS

  v8f acc[4] = {};
#pragma unroll 4
  for (int kk = 0; kk < 16; ++kk) {
    const int koff = kk * 4 + half * 2;          // even
    const int pr = (kk << 1) + half;             // pair-row in ldsM2
    const v2f a = *(const v2f*)(mybuf + cl * 68 + koff);  // A(16x4) fragment
#pragma unroll
    for (int n = 0; n < 4; ++n) {
      const int mcol = n * 16 + cl;
      const v2f bf = *(const v2f*)(ldsM2 + pr * 130 + 2 * mcol);
      acc[n] = wmma4(a, bf, acc[n]);
    }
  }
  // epilogue: broadcast adds, stage tile to LDS
  const float* tiw = ws + WS_TIW + (b * L + i) * 64;
  const float* bjw = ws + WS_BJW + (b * L + j0) * 64;
  float* ost = &outst[wave][0];
#pragma unroll
  for (int n = 0; n < 4; ++n) {
    const int col = n * 16 + cl;
    const float tadd = tiw[col] + ldsCw[col];
#pragma unroll
    for (int v = 0; v < 8; ++v) {
      const int M = v + half * 8;
      ost[M * 68 + col] = acc[n][v] + tadd + bjw[M * 64 + col];
    }
  }
  wait_ds0();  // tile staged: async store engine may now read our LDS region
  // DMA the 4 KB output tile back to global (contiguous, full-line stores)
  float* gdst = new_edge + (((size_t)(b * L + i)) * L + j0) * DE;
#pragma unroll
  for (int t = 0; t < 8; ++t) {
    const int idx = t * 32 + lane;
    const int row = idx >> 4, c4 = idx & 15;
    async_store_b128(gdst + idx * 4, ost + row * 68 + c4 * 4);
  }
  // fused edge-logit matvecs while the stores drain:
  // lanes<16 -> s_e_t, lanes>=16 -> s_e_b (transposed store)
  {
    const int row = cl;
    const float* Av = half ? ldsAb : ldsAt;
    const float* cadd = half ? ldsCb : ldsCt;
    const float* orow = ost + row * 68;
    float* se = ws + (half ? WS_SEB : WS_SET);
    const size_t base = half ? (((size_t)(b * L + (j0 + row))) * L + i) * H
                             : (((size_t)(b * L + i)) * L + (j0 + row)) * H;
#pragma unroll
    for (int h = 0; h < H; ++h) {
      float s = cadd[h];
      for (int k = 0; k < 64; ++k) s += orow[k] * Av[k * 8 + h];
      se[base + h] = s;
    }
  }
  // S_ENDPGM performs an implicit wait-idle, covering outstanding ASYNCcnt.
}

// ============ K3/K4: attention (two-pass softmax + WMMA P@V) ============
__global__ __launch_bounds__(256) void attn_kernel(
    const float* __restrict__ q, const float* __restrict__ kc,
    const float* __restrict__ se, const float* __restrict__ V,
    float* __restrict__ agg) {
  const int tid = threadIdx.x, wave = tid >> 5, lane = tid & 31;
  const int half = lane >> 4, row = lane & 15;
  const int tile = blockIdx.x * 8 + wave;  // 512 tiles: (b,h,i-tile)
  const int b = tile >> 8;
  const int rem = tile & 255;
  const int h = rem >> 5;
  const int i0 = (rem & 31) << 4;
  const int i = i0 + row;

  const float qv = q[(b * L + i) * H + h];
  const float* kch = kc + (size_t)b * L * H + h;
  const float* seh = se + (((size_t)(b * L + i)) * L) * H + h;

  // pass 1: online max & sumexp; each lane scans half the j-range for its row
  float m = -3.4e38f, sum = 0.f;
  for (int jj = 0; jj < 256; ++jj) {
    const int j = half * 256 + jj;
    float s = leaky(qv + kch[(size_t)j * H] + seh[(size_t)j * H]);
    if (s > m) {
      sum = sum * __expf(m - s) + 1.f;
      m = s;
    } else {
      sum += __expf(s - m);
    }
  }
  float mo = __shfl_xor(m, 16, 32);
  float so = __shfl_xor(sum, 16, 32);
  float mn = fmaxf(m, mo);
  sum = sum * __expf(m - mn) + so * __expf(mo - mn);
  m = mn;
  const float rs = 1.f / sum;

  // pass 2: P(16x512) @ V(512x32) via WMMA f32 16x16x4, A-frags on the fly
  v8f acc0 = {}, acc1 = {};
  const float* Vh = V + ((size_t)b * L * H + h) * DK;
  for (int kk = 0; kk < 128; ++kk) {
    const int jA = kk * 4 + half * 2;
    float s0 = leaky(qv + kch[(size_t)jA * H] + seh[(size_t)jA * H]);
    float s1 = leaky(qv + kch[(size_t)(jA + 1) * H] + seh[(size_t)(jA + 1) * H]);
    v2f a;
    a.x = __expf(s0 - m) * rs;
    a.y = __expf(s1 - m) * rs;
    v2f b0, b1;
    b0.x = Vh[(size_t)jA * H * DK + row];
    b0.y = Vh[(size_t)(jA + 1) * H * DK + row];
    b1.x = Vh[(size_t)jA * H * DK + 16 + row];
    b1.y = Vh[(size_t)(jA + 1) * H * DK + 16 + row];
    acc0 = wmma4(a, b0, acc0);
    acc1 = wmma4(a, b1, acc1);
  }
#pragma unroll
  for (int v = 0; v < 8; ++v) {
    const int M = v + half * 8;
    float* dst = agg + ((size_t)(b * L + i0 + M) * H + h) * DK;
    dst[row] = acc0[v];
    dst[16 + row] = acc1[v];
  }
}

// ============ K5: residual + leaky + output GEMV ============
__global__ __launch_bounds__(256) void out_kernel(
    const float* __restrict__ ws, const float* __restrict__ out_n_w,
    const float* __restrict__ out_n_b, float* __restrict__ out) {
  const bool isT = blockIdx.x < (unsigned)(BATCH * L);
  const int r = isT ? blockIdx.x : blockIdx.x - BATCH * L;
  const int tid = threadIdx.x;
  __shared__ float xv[256];
  const float p2 = ws[(isT ? WS_TP2 : WS_BP2) + r * DN + tid];
  const float ag = ws[(isT ? WS_AGGT : WS_AGGB) + r * DN + tid];
  xv[tid] = p2 + leaky(ag);
  __syncthreads();
  float s = out_n_b[tid];
  for (int k = 0; k < DN; ++k) s += xv[k] * out_n_w[k * DN + tid];
  out[(isT ? 0 : BATCH * L * DN) + r * DN + tid] = s;
}

extern "C" void kernel_launch(void* const* d_in, const int* in_sizes, int n_in,
                              void* d_out, int out_size, void* d_ws, size_t ws_size,
                              hipStream_t stream) {
  const float* target   = (const float*)d_in[0];
  const float* binder   = (const float*)d_in[1];
  const float* edge     = (const float*)d_in[2];
  const float* diff     = (const float*)d_in[3];
  const float* Wn_w     = (const float*)d_in[4];
  const float* Wn_b     = (const float*)d_in[5];
  const float* Wh_w     = (const float*)d_in[6];
  const float* Wh_b     = (const float*)d_in[7];
  const float* We_w     = (const float*)d_in[8];
  const float* We_b     = (const float*)d_in[9];
  const float* Wn2_w    = (const float*)d_in[10];
  const float* Wn2_b    = (const float*)d_in[11];
  const float* We2_w    = (const float*)d_in[12];
  const float* We2_b    = (const float*)d_in[13];
  const float* Wd_w     = (const float*)d_in[14];
  const float* Wd_b     = (const float*)d_in[15];
  const float* Wd2_w    = (const float*)d_in[16];
  const float* Wd2_b    = (const float*)d_in[17];
  const float* attn_t_w = (const float*)d_in[18];
  const float* attn_b_w = (const float*)d_in[19];
  const float* edge_w   = (const float*)d_in[20];
  const float* edge_b   = (const float*)d_in[21];
  const float* out_n_w  = (const float*)d_in[22];
  const float* out_n_b  = (const float*)d_in[23];
  const float* out_e_w  = (const float*)d_in[24];
  const float* out_e_b  = (const float*)d_in[25];
  float* ws = (float*)d_ws;
  float* out = (float*)d_out;

  setup_kernel<<<1, 64, 0, stream>>>(We_w, We_b, We2_w, We2_b, edge_w, edge_b,
                                     out_e_w, out_e_b, attn_t_w, attn_b_w, ws);
  node_kernel<<<2 * BATCH * L, 256, 0, stream>>>(
      target, binder, diff, Wn_w, Wn_b, Wn2_w, Wn2_b, Wd_w, Wd_b, Wd2_w, Wd2_b,
      Wh_w, Wh_b, edge_w, out_e_w, attn_t_w, attn_b_w, ws);
  edge_kernel<<<(BATCH * L * L / 16) / 8, 256, 0, stream>>>(
      edge, ws, out + 2 * BATCH * L * DN);
  attn_kernel<<<(BATCH * H * (L / 16)) / 8, 256, 0, stream>>>(
      ws + WS_QT, ws + WS_KT, ws + WS_SET, ws + WS_WHB, ws + WS_AGGT);
  attn_kernel<<<(BATCH * H * (L / 16)) / 8, 256, 0, stream>>>(
      ws + WS_QB, ws + WS_KB, ws + WS_SEB, ws + WS_WHT, ws + WS_AGGB);
  out_kernel<<<2 * BATCH * L, 256, 0, stream>>>(ws, out_n_w, out_n_b, out);
}